// GAT_4904852652497
// MI455X (gfx1250) — compile-verified
//
#include <hip/hip_runtime.h>
#include <math.h>

// ---------------------------------------------------------------------------
// 2-layer GAT for MI455X (gfx1250).
// Dense projections: V_WMMA_F32_16X16X4_F32 with the weight tile staged into
// LDS via GLOBAL_LOAD_ASYNC_TO_LDS_B128 (ASYNCcnt) and consumed via ds_load.
// Edge softmax/aggregation phases are L2-resident scatter/gather (all node
// tables total ~163 MB < 192 MB L2).
// ---------------------------------------------------------------------------

typedef __attribute__((ext_vector_type(2))) float v2f;
typedef __attribute__((ext_vector_type(8))) float v8f;

#define GAT_NEG_SLOPE 0.2f
#define GAT_K 128            // inner dim is 128 for both layers
#define GAT_TN 64            // B-tile width per block (4 waves x 16 cols)

// ---------------- float atomic max (sign-split trick, init = -inf) ---------
__device__ __forceinline__ void atomicMaxF(float* addr, float v) {
    if (v >= 0.0f) {
        atomicMax((int*)addr, __float_as_int(v));
    } else {
        atomicMin((unsigned int*)addr, __float_as_uint(v));
    }
}

// ---------------- dense GEMM: C[N x NC] = A[N x K] * B[K x NC], K == 128 ---
// Block = 128 threads (4 waves); block tile = 16 rows x 64 cols.
// Stage B[0:128, nb:nb+64] (32 KB) into LDS with async b128 copies, then each
// wave runs a 16x16 WMMA tile; B fragments come from LDS (ds_load).
// A-fragment (16x4 f32): lanes 0-15 hold K = k+0,k+1; lanes 16-31 hold k+2,k+3.
// C/D layout: VGPR i -> row (lane<16 ? i : 8+i), col = lane & 15.
__global__ void gat_gemm_wmma_f32(const float* __restrict__ A,
                                  const float* __restrict__ B,
                                  float* __restrict__ C,
                                  int NC) {
    __shared__ float bs[GAT_K * GAT_TN];          // 32 KB

    const int tid  = threadIdx.x;                 // 0..127
    const int wave = tid >> 5;
    const int lane = tid & 31;
    const int m0   = blockIdx.x * 16;
    const int nb   = blockIdx.y * GAT_TN;

    // ---- async-copy B tile: 128 rows x 64 cols = 2048 float4s, 16/thread --
    for (int it = 0; it < 16; ++it) {
        int elt = it * 128 + tid;                 // 0..2047
        int row = elt >> 4;                       // 16 float4s per row
        int c4  = elt & 15;
        const float* gp = B + (size_t)row * NC + nb + c4 * 4;
        unsigned lofs = (unsigned)(uintptr_t)(const void*)&bs[row * GAT_TN + c4 * 4];
        asm volatile("global_load_async_to_lds_b128 %0, %1, off"
                     :: "v"(lofs), "v"(gp) : "memory");
    }
    asm volatile("s_wait_asynccnt 0x0" ::: "memory");
    __syncthreads();

    // ---- per-wave 16x16 WMMA tile -----------------------------------------
    const int r   = lane & 15;                    // row (A) / col (B,C)
    const int kh  = (lane >> 4) << 1;             // 0 or 2: K sub-offset
    const int ncl = wave * 16 + r;                // col within LDS tile

    v8f acc = {};
    const float* ap = A + (size_t)(m0 + r) * GAT_K + kh;
    const float* bl = &bs[kh * GAT_TN + ncl];

#pragma unroll 8
    for (int k = 0; k < GAT_K; k += 4) {
        v2f a, b;
        a[0] = ap[0];
        a[1] = ap[1];
        b[0] = bl[0];                              // ds_load
        b[1] = bl[GAT_TN];
        acc = __builtin_amdgcn_wmma_f32_16x16x4_f32(
            /*neg_a=*/false, a, /*neg_b=*/false, b,
            /*c_mod=*/(short)0, acc, /*reuse_a=*/false, /*reuse_b=*/false);
        ap += 4;
        bl += 4 * GAT_TN;
    }

    const int mb = (lane >> 4) << 3;              // 0 or 8
    float* cp = C + (size_t)(m0 + mb) * NC + nb + wave * 16 + r;
#pragma unroll
    for (int i = 0; i < 8; ++i) cp[(size_t)i * NC] = acc[i];
}

// -------- per-(node,head) attention coefficients: <h, a_src>, <h, a_dst> ---
__global__ void gat_alpha(const float* __restrict__ h,
                          const float* __restrict__ a_src,
                          const float* __restrict__ a_dst,
                          float* __restrict__ asrc,
                          float* __restrict__ adst,
                          int N, int H, int C) {
    int i = blockIdx.x * blockDim.x + threadIdx.x;
    if (i >= N * H) return;
    int n  = i / H;
    int hh = i - n * H;
    const float* hp = h + (size_t)n * H * C + (size_t)hh * C;
    const float* sp = a_src + (size_t)hh * C;
    const float* dp = a_dst + (size_t)hh * C;
    float vs = 0.0f, vd = 0.0f;
    for (int c = 0; c < C; c += 4) {
        float4 hv = *(const float4*)(hp + c);
        float4 sv = *(const float4*)(sp + c);
        float4 dv = *(const float4*)(dp + c);
        vs += hv.x * sv.x + hv.y * sv.y + hv.z * sv.z + hv.w * sv.w;
        vd += hv.x * dv.x + hv.y * dv.y + hv.z * dv.z + hv.w * dv.w;
    }
    asrc[i] = vs;
    adst[i] = vd;
}

// ---------------- buffer fill ----------------------------------------------
__global__ void gat_fill(float* __restrict__ p, float v, int n) {
    int i = blockIdx.x * blockDim.x + threadIdx.x;
    if (i < n) p[i] = v;
}

__device__ __forceinline__ void edge_nodes(const int* __restrict__ ei,
                                           int E, int e, int& s, int& d) {
    if (e < E) { s = ei[e]; d = ei[E + e]; }
    else       { s = e - E; d = s; }       // self loop
}

__device__ __forceinline__ float gat_logit(const float* __restrict__ asrc,
                                           const float* __restrict__ adst,
                                           int s, int d, int H, int hh) {
    float x = asrc[s * H + hh] + adst[d * H + hh];
    return x > 0.0f ? x : GAT_NEG_SLOPE * x;
}

// ---------------- edge pass 1: segment max --------------------------------
__global__ void gat_edge_max(const int* __restrict__ ei, int E, int N, int H,
                             const float* __restrict__ asrc,
                             const float* __restrict__ adst,
                             float* __restrict__ m) {
    int idx = blockIdx.x * blockDim.x + threadIdx.x;
    if (idx >= (E + N) * H) return;
    int e  = idx / H;
    int hh = idx - e * H;
    int s, d;
    edge_nodes(ei, E, e, s, d);
    float x = gat_logit(asrc, adst, s, d, H, hh);
    atomicMaxF(&m[d * H + hh], x);
}

// ---------------- edge pass 2: segment exp-sum -----------------------------
__global__ void gat_edge_expsum(const int* __restrict__ ei, int E, int N, int H,
                                const float* __restrict__ asrc,
                                const float* __restrict__ adst,
                                const float* __restrict__ m,
                                float* __restrict__ ssum) {
    int idx = blockIdx.x * blockDim.x + threadIdx.x;
    if (idx >= (E + N) * H) return;
    int e  = idx / H;
    int hh = idx - e * H;
    int s, d;
    edge_nodes(ei, E, e, s, d);
    float x = gat_logit(asrc, adst, s, d, H, hh);
    atomicAdd(&ssum[d * H + hh], __expf(x - m[d * H + hh]));
}

// ---------------- edge pass 3: weighted message aggregation ----------------
// Thread = (edge, head); gathers h[src] as b128 loads, scatter-adds to acc[dst].
__global__ void gat_edge_aggregate(const int* __restrict__ ei, int E, int N,
                                   int H, int C,
                                   const float* __restrict__ asrc,
                                   const float* __restrict__ adst,
                                   const float* __restrict__ m,
                                   const float* __restrict__ ssum,
                                   const float* __restrict__ h,
                                   float* __restrict__ acc) {
    int idx = blockIdx.x * blockDim.x + threadIdx.x;
    if (idx >= (E + N) * H) return;
    int e  = idx / H;
    int hh = idx - e * H;
    int s, d;
    edge_nodes(ei, E, e, s, d);
    float x    = gat_logit(asrc, adst, s, d, H, hh);
    float coef = __expf(x - m[d * H + hh]) / ssum[d * H + hh];
    const float* hp = h   + (size_t)s * H * C + (size_t)hh * C;
    float*       op = acc + (size_t)d * H * C + (size_t)hh * C;
    for (int c = 0; c < C; c += 4) {
        float4 v = *(const float4*)(hp + c);
        atomicAdd(op + c + 0, v.x * coef);
        atomicAdd(op + c + 1, v.y * coef);
        atomicAdd(op + c + 2, v.z * coef);
        atomicAdd(op + c + 3, v.w * coef);
    }
}

// ---------------- layer-1 epilogue: +bias, ELU (in place) ------------------
__global__ void gat_bias_elu(float* __restrict__ acc,
                             const float* __restrict__ b, int N, int F) {
    int i = blockIdx.x * blockDim.x + threadIdx.x;
    if (i >= N * F) return;
    int j = i % F;
    float x = acc[i] + b[j];
    acc[i] = x > 0.0f ? x : (__expf(x) - 1.0f);
}

// ---------------- layer-2 epilogue: mean over heads, +bias -----------------
__global__ void gat_head_mean(const float* __restrict__ acc,
                              const float* __restrict__ b,
                              float* __restrict__ out,
                              int N, int H, int C) {
    int i = blockIdx.x * blockDim.x + threadIdx.x;
    if (i >= N * C) return;
    int n = i / C;
    int c = i - n * C;
    float v = 0.0f;
    for (int hh = 0; hh < H; ++hh) v += acc[(size_t)n * H * C + (size_t)hh * C + c];
    out[i] = v * (1.0f / (float)H) + b[c];
}

// ---------------------------------------------------------------------------
extern "C" void kernel_launch(void* const* d_in, const int* in_sizes, int n_in,
                              void* d_out, int out_size, void* d_ws, size_t ws_size,
                              hipStream_t stream) {
    const float* x   = (const float*)d_in[0];
    const int*   ei  = (const int*)d_in[1];      // edge_index, [2, E] flat
    const float* W1  = (const float*)d_in[2];
    const float* as1 = (const float*)d_in[3];
    const float* ad1 = (const float*)d_in[4];
    const float* b1  = (const float*)d_in[5];
    const float* W2  = (const float*)d_in[6];
    const float* as2 = (const float*)d_in[7];
    const float* ad2 = (const float*)d_in[8];
    const float* b2  = (const float*)d_in[9];
    float* out = (float*)d_out;
    (void)n_in; (void)out_size; (void)ws_size;

    const int IN = 128;
    const int N  = in_sizes[0] / IN;     // 50000 (multiple of 16)
    const int E  = in_sizes[1] / 2;      // 800000
    const int H1 = 4, H2 = 8, C1 = 32, C2 = 32;
    const int F1 = H1 * C1;              // 128
    const int F2 = H2 * C2;              // 256
    const int ET = E + N;                // edges incl. self loops

    // Workspace layout (floats). Total ~40.8M floats (~163 MB).
    float* ws = (float*)d_ws;
    size_t o = 0;
    float* h1   = ws + o; o += (size_t)N * F1;
    float* acc1 = ws + o; o += (size_t)N * F1;   // becomes ELU output in place
    float* h2   = ws + o; o += (size_t)N * F2;
    float* acc2 = ws + o; o += (size_t)N * F2;
    float* v_s1 = ws + o; o += (size_t)N * H1;
    float* v_d1 = ws + o; o += (size_t)N * H1;
    float* m1   = ws + o; o += (size_t)N * H1;
    float* s1   = ws + o; o += (size_t)N * H1;
    float* v_s2 = ws + o; o += (size_t)N * H2;
    float* v_d2 = ws + o; o += (size_t)N * H2;
    float* m2   = ws + o; o += (size_t)N * H2;
    float* s2   = ws + o; o += (size_t)N * H2;

    const int TB = 256;
    auto gr = [](long long n, int b) { return (unsigned)((n + b - 1) / b); };

    // ================= Layer 1 =================
    gat_gemm_wmma_f32<<<dim3(N / 16, F1 / GAT_TN), 128, 0, stream>>>(x, W1, h1, F1);
    gat_alpha<<<gr((long long)N * H1, TB), TB, 0, stream>>>(h1, as1, ad1, v_s1, v_d1, N, H1, C1);

    gat_fill<<<gr((long long)N * H1, TB), TB, 0, stream>>>(m1, -INFINITY, N * H1);
    gat_fill<<<gr((long long)N * H1, TB), TB, 0, stream>>>(s1, 0.0f, N * H1);
    gat_fill<<<gr((long long)N * F1, TB), TB, 0, stream>>>(acc1, 0.0f, N * F1);

    gat_edge_max      <<<gr((long long)ET * H1, TB), TB, 0, stream>>>(ei, E, N, H1, v_s1, v_d1, m1);
    gat_edge_expsum   <<<gr((long long)ET * H1, TB), TB, 0, stream>>>(ei, E, N, H1, v_s1, v_d1, m1, s1);
    gat_edge_aggregate<<<gr((long long)ET * H1, TB), TB, 0, stream>>>(ei, E, N, H1, C1, v_s1, v_d1, m1, s1, h1, acc1);

    gat_bias_elu<<<gr((long long)N * F1, TB), TB, 0, stream>>>(acc1, b1, N, F1);

    // ================= Layer 2 =================
    gat_gemm_wmma_f32<<<dim3(N / 16, F2 / GAT_TN), 128, 0, stream>>>(acc1, W2, h2, F2);
    gat_alpha<<<gr((long long)N * H2, TB), TB, 0, stream>>>(h2, as2, ad2, v_s2, v_d2, N, H2, C2);

    gat_fill<<<gr((long long)N * H2, TB), TB, 0, stream>>>(m2, -INFINITY, N * H2);
    gat_fill<<<gr((long long)N * H2, TB), TB, 0, stream>>>(s2, 0.0f, N * H2);
    gat_fill<<<gr((long long)N * F2, TB), TB, 0, stream>>>(acc2, 0.0f, N * F2);

    gat_edge_max      <<<gr((long long)ET * H2, TB), TB, 0, stream>>>(ei, E, N, H2, v_s2, v_d2, m2);
    gat_edge_expsum   <<<gr((long long)ET * H2, TB), TB, 0, stream>>>(ei, E, N, H2, v_s2, v_d2, m2, s2);
    gat_edge_aggregate<<<gr((long long)ET * H2, TB), TB, 0, stream>>>(ei, E, N, H2, C2, v_s2, v_d2, m2, s2, h2, acc2);

    gat_head_mean<<<gr((long long)N * C2, TB), TB, 0, stream>>>(acc2, b2, out, N, H2, C2);
}